// RNN_43396349559294
// MI455X (gfx1250) — compile-verified
//
#include <hip/hip_runtime.h>
#include <hip/hip_bf16.h>
#include <stdint.h>

#define B_ 32
#define S_ 2048
#define I_ 256
#define H_ 512

typedef __attribute__((ext_vector_type(2))) float    v2f;
typedef __attribute__((ext_vector_type(8))) float    v8f;
typedef __attribute__((ext_vector_type(4))) unsigned v4u;
typedef __attribute__((ext_vector_type(4))) int      v4i_t;
typedef __attribute__((ext_vector_type(8))) int      v8i_t;

#ifndef __has_builtin
#define __has_builtin(x) 0
#endif
#if __has_builtin(__builtin_amdgcn_tensor_load_to_lds) && __has_builtin(__builtin_amdgcn_s_wait_tensorcnt)
#define HAS_TDM 1
#else
#define HAS_TDM 0
#endif

// ---------------------------------------------------------------------------
// f32 WMMA: D(16x16) = A(16x4) * B(4x16) + C   (full fp32, matches reference)
// A frag (per lane): m = lane&15, k0 = kbase + 2*(lane>>4); holds {A[m][k0],A[m][k0+1]}
// B frag (per lane): n = lane&15, k0 = kbase + 2*(lane>>4); holds {B[k0][n],B[k0+1][n]}
// C/D: vgpr r, lanes0-15 -> M=r, lanes16-31 -> M=r+8; N = lane&15
// ---------------------------------------------------------------------------
__device__ __forceinline__ v8f wmma4(v2f a, v2f b, v8f c) {
  return __builtin_amdgcn_wmma_f32_16x16x4_f32(false, a, false, b, (short)0, c,
                                               false, false);
}

#if HAS_TDM
// Build a 2D TDM descriptor (D#) per CDNA5 ISA ch.8 and issue tensor_load_to_lds.
// This toolchain exposes the 6-arg builtin:
//   (uint32x4 g0, int32x8 g1, int32x4 g2, int32x4 g3, int32x8 extra, i32 cpol)
// dims/strides in 4-byte elements; optional LDS padding: pad_amt dwords every
// pad_int dwords (codes: interval 0=2..7=256 ; amount 0=1..127=128).
__device__ __forceinline__ void tdm_load_2d(unsigned lds_byte_off,
                                            const void* gptr,
                                            unsigned tdim0, unsigned tdim1,
                                            unsigned tile0, unsigned tile1,
                                            unsigned stride0,
                                            unsigned pad_int_code,
                                            unsigned pad_amt_code,
                                            bool pad_en) {
  unsigned long long ga = (unsigned long long)(uintptr_t)gptr;
  v4u g0;
  g0.x = 1u;                                      // count=1, user descriptor
  g0.y = lds_byte_off;                            // lds_addr (bytes, wg-relative)
  g0.z = (unsigned)(ga & 0xFFFFFFFFull);          // global_addr[31:0]
  g0.w = (unsigned)((ga >> 32) & 0x1FFFFFFull) | (2u << 30);  // addr[56:32] | type=2
  unsigned d0 = (2u << 16);                       // data_size = 4B, no multicast
  if (pad_en) d0 |= (1u << 20) | (pad_int_code << 22) | (pad_amt_code << 25);
  v8i_t g1;
  g1[0] = (int)d0;
  g1[1] = (int)((tdim0 & 0xFFFFu) << 16);                         // tensor_dim0[15:0]
  g1[2] = (int)(((tdim0 >> 16) & 0xFFFFu) | ((tdim1 & 0xFFFFu) << 16));
  g1[3] = (int)(((tdim1 >> 16) & 0xFFFFu) | ((tile0 & 0xFFFFu) << 16));
  g1[4] = (int)(tile1 & 0xFFFFu);                 // tile_dim1; tile_dim2=0 (2D)
  g1[5] = (int)stride0;                           // tensor_dim0_stride[31:0]
  g1[6] = 0;                                      // stride0[47:32]=0, stride1 lo=0
  g1[7] = 0;
  v4i_t z4 = {0, 0, 0, 0};                        // groups 2/3 unused (<=2D)
  v8i_t z8 = {0, 0, 0, 0, 0, 0, 0, 0};
  __builtin_amdgcn_tensor_load_to_lds(g0, g1, z4, z4, z8, 0);
}
#endif

// padded LDS index for a [rows][512] tile staged with "pad 2 every 256 dwords"
// -> effective pitch 516 with a 2-dword pad in the middle. Keeps all WMMA
// fragment b64 loads 8B-aligned and (pitch%64==4) bank-conflict free.
__device__ __forceinline__ int lidx516(int r, int c) {
  return r * 516 + c + ((c >= 256) ? 2 : 0);
}

// ---------------------------------------------------------------------------
// Kernel 1: Z[65536,512] = X[65536,256] @ Wih^T + b_ih   (written into d_out)
// 128x64 tile per WG, 8 waves as 4x2, each wave 32x32 via 2x2 WMMA accums.
// ---------------------------------------------------------------------------
__global__ void rnn_zin_gemm(const float* __restrict__ X,
                             const float* __restrict__ Wih,
                             const float* __restrict__ bih,
                             float* __restrict__ Z) {
  extern __shared__ float smem[];
  float* sX = smem;              // 128 x 36 (pitch 36 via TDM pad 4-per-32)
  float* sW = smem + 128 * 36;   // 64 x 36
  const int tid  = threadIdx.x;
  const int lane = tid & 31;
  const int w    = tid >> 5;     // 0..7
  const int wm   = w >> 1;       // 0..3
  const int wn   = w & 1;        // 0..1
  const int bm   = blockIdx.y * 128;
  const int bn   = blockIdx.x * 64;
  const int ml   = lane & 15;
  const int kh   = 2 * (lane >> 4);
  const int hh   = lane >> 4;

  v8f acc[2][2] = {};

  for (int kb = 0; kb < I_; kb += 32) {
    __syncthreads();
#if HAS_TDM
    if (tid < 32) {  // wave 0 drives both DMAs, then drains TENSORcnt
      tdm_load_2d(0u, X + (size_t)bm * I_ + kb, 32, 128, 32, 128, I_, 4u, 3u, true);
      tdm_load_2d(128u * 36u * 4u, Wih + (size_t)bn * I_ + kb, 32, 64, 32, 64, I_,
                  4u, 3u, true);
      __builtin_amdgcn_s_wait_tensorcnt(0);
    }
#else
    for (int i = tid; i < 128 * 32; i += 256) {
      int r = i >> 5, c = i & 31;
      sX[r * 36 + c] = X[(size_t)(bm + r) * I_ + kb + c];
    }
    for (int i = tid; i < 64 * 32; i += 256) {
      int r = i >> 5, c = i & 31;
      sW[r * 36 + c] = Wih[(size_t)(bn + r) * I_ + kb + c];
    }
#endif
    __syncthreads();

#pragma unroll
    for (int ks = 0; ks < 8; ++ks) {
      const int k = ks * 4 + kh;
      v2f a0 = *(const v2f*)&sX[(wm * 32 + ml) * 36 + k];
      v2f a1 = *(const v2f*)&sX[(wm * 32 + 16 + ml) * 36 + k];
      v2f b0 = *(const v2f*)&sW[(wn * 32 + ml) * 36 + k];
      v2f b1 = *(const v2f*)&sW[(wn * 32 + 16 + ml) * 36 + k];
      acc[0][0] = wmma4(a0, b0, acc[0][0]);
      acc[0][1] = wmma4(a0, b1, acc[0][1]);
      acc[1][0] = wmma4(a1, b0, acc[1][0]);
      acc[1][1] = wmma4(a1, b1, acc[1][1]);
    }
  }

#pragma unroll
  for (int nt = 0; nt < 2; ++nt) {
    const int col  = bn + wn * 32 + nt * 16 + ml;
    const float bb = bih[col];
#pragma unroll
    for (int mt = 0; mt < 2; ++mt) {
#pragma unroll
      for (int r = 0; r < 8; ++r) {
        const int row = bm + wm * 32 + mt * 16 + r + 8 * hh;
        Z[(size_t)row * H_ + col] = acc[mt][nt][r] + bb;
      }
    }
  }
}

// ---------------------------------------------------------------------------
// Init: zero the arrive counter, seed h ping-pong buffer 0 with h0.
// (d_ws is poisoned before timing; re-init every launch for determinism.)
// ---------------------------------------------------------------------------
__global__ void rnn_init(const float* __restrict__ h0, float* __restrict__ wsf,
                         unsigned* __restrict__ cnt) {
  const int i = blockIdx.x * blockDim.x + threadIdx.x;  // exactly 16384 threads
  if (i == 0) *cnt = 0u;
  wsf[1024 + i] = h0[i];  // hbuf0 at byte offset 4096
}

// ---------------------------------------------------------------------------
// Kernel 2: sequential scan. 16 WGs, WG g owns features [32g, 32g+32).
// W_hh slice (32x512, 64KB) resident in LDS for all 2048 steps.
// Per step: TDM-stage h_{t-1} (64KB) from ping-pong L2 buffer, 8 waves split
// K=512, WMMA 32x32 partials, fixed-order LDS reduction, relu epilogue,
// write h_t to d_out (overwriting z_t in place) + other ping-pong buffer,
// device-scope arrive-counter barrier (plus s_cluster_barrier, NOP unclustered).
// ---------------------------------------------------------------------------
__global__ void rnn_scan(const float* __restrict__ Whh,
                         const float* __restrict__ bhh,
                         float* __restrict__ Out,
                         float* __restrict__ wsf,
                         unsigned* __restrict__ cnt) {
  extern __shared__ float smem[];
  float* sW   = smem;                 // 32 x 516 padded  (66,048 B)
  float* sH   = smem + 16512;         // 32 x 516 padded
  float* sRed = smem + 2 * 16512;     // 8 slabs x 32 x 32 (32,768 B)

  const int tid   = threadIdx.x;
  const int lane  = tid & 31;
  const int w     = tid >> 5;         // 0..7, K-chunk owner
  const int fbase = blockIdx.x * 32;
  const int ml    = lane & 15;
  const int kh    = 2 * (lane >> 4);
  const int hh    = lane >> 4;
  const unsigned nwg = gridDim.x;     // 16

  float* hb0 = wsf + 1024;            // ping-pong h buffers, 64KB each (L2-hot)
  float* hb1 = wsf + 1024 + 16384;

  // Stage the W_hh slice once (rows = our 32 output features).
#if HAS_TDM
  if (tid < 32) {
    tdm_load_2d(0u, Whh + (size_t)fbase * H_, H_, 32, H_, 32, H_, 7u, 1u, true);
    __builtin_amdgcn_s_wait_tensorcnt(0);
  }
#else
  for (int i = tid; i < 32 * H_; i += 256) {
    const int r = i >> 9, c = i & 511;
    sW[lidx516(r, c)] = Whh[(size_t)(fbase + r) * H_ + c];
  }
#endif

  for (int t = 0; t < S_; ++t) {
    const float* hcur = (t & 1) ? hb1 : hb0;
    float* hnext      = (t & 1) ? hb0 : hb1;

    // Stage h_{t-1}: 32x512 -> padded LDS (pitch 516).
#if HAS_TDM
    if (tid < 32) {
      tdm_load_2d(16512u * 4u, hcur, H_, 32, H_, 32, H_, 7u, 1u, true);
      __builtin_amdgcn_s_wait_tensorcnt(0);
    }
#else
    for (int i = tid; i < 32 * H_; i += 256) {
      const int r = i >> 9, c = i & 511;
      sH[lidx516(r, c)] = hcur[(size_t)r * H_ + c];
    }
#endif
    __syncthreads();  // sW ready (t==0) and sH ready

    // y[32b x 32f] partials: wave w covers K in [64w, 64w+64).
    v8f acc[2][2] = {};
#pragma unroll 4
    for (int ks = 0; ks < 16; ++ks) {
      const int k  = w * 64 + ks * 4 + kh;
      const int kp = k + ((k >= 256) ? 2 : 0);
      v2f a0 = *(const v2f*)&sH[ml * 516 + kp];          // h[m][k..k+1]
      v2f a1 = *(const v2f*)&sH[(16 + ml) * 516 + kp];
      v2f b0 = *(const v2f*)&sW[ml * 516 + kp];          // W[n][k..k+1]
      v2f b1 = *(const v2f*)&sW[(16 + ml) * 516 + kp];
      acc[0][0] = wmma4(a0, b0, acc[0][0]);
      acc[0][1] = wmma4(a0, b1, acc[0][1]);
      acc[1][0] = wmma4(a1, b0, acc[1][0]);
      acc[1][1] = wmma4(a1, b1, acc[1][1]);
    }

    // Dump per-wave partials; reduce in fixed order (deterministic FP).
#pragma unroll
    for (int mt = 0; mt < 2; ++mt)
#pragma unroll
      for (int nt = 0; nt < 2; ++nt)
#pragma unroll
        for (int r = 0; r < 8; ++r)
          sRed[w * 1024 + (mt * 16 + r + 8 * hh) * 32 + nt * 16 + ml] =
              acc[mt][nt][r];
    __syncthreads();

    {
      const int e = tid * 4, m = e >> 5, n = e & 31;  // 256 thr x float4 = 32x32
      float s0 = 0.f, s1 = 0.f, s2 = 0.f, s3 = 0.f;
#pragma unroll
      for (int ww = 0; ww < 8; ++ww) {
        const float4 p = *(const float4*)&sRed[ww * 1024 + m * 32 + n];
        s0 += p.x; s1 += p.y; s2 += p.z; s3 += p.w;
      }
      const size_t zoff = ((size_t)m * S_ + t) * H_ + fbase + n;
      const float4 z  = *(const float4*)&Out[zoff];      // z_t (has b_ih)
      const float4 bb = *(const float4*)&bhh[fbase + n];
      float4 h;
      h.x = fmaxf(z.x + bb.x + s0, 0.f);
      h.y = fmaxf(z.y + bb.y + s1, 0.f);
      h.z = fmaxf(z.z + bb.z + s2, 0.f);
      h.w = fmaxf(z.w + bb.w + s3, 0.f);
      *(float4*)&Out[zoff] = h;                          // overwrite z with h_t
      *(float4*)&hnext[m * H_ + fbase + n] = h;          // publish for all WGs
      if (t == S_ - 1)
        *(float4*)&Out[(size_t)B_ * S_ * H_ + (size_t)m * H_ + fbase + n] = h;
      if ((t + 1 < S_) && (n == 0))                      // warm L2/L0 for z_{t+1}
        __builtin_prefetch(&Out[((size_t)m * S_ + t + 1) * H_ + fbase], 0, 3);
    }

    // --- inter-WG step barrier (ping-pong => one barrier/step suffices) ----
    __threadfence();    // make hnext stores device-visible
    __syncthreads();
    if (tid == 0) {
      __hip_atomic_fetch_add(cnt, 1u, __ATOMIC_RELEASE, __HIP_MEMORY_SCOPE_AGENT);
      const unsigned target = nwg * (unsigned)(t + 1);
      while (__hip_atomic_load(cnt, __ATOMIC_ACQUIRE, __HIP_MEMORY_SCOPE_AGENT) <
             target)
        __builtin_amdgcn_s_sleep(8);
    }
    if (tid < 32)
      __builtin_amdgcn_s_cluster_barrier();  // NOP unless launched as a cluster
    __syncthreads();
    __threadfence();    // acquire: invalidate stale near caches before next read
  }
}

// ---------------------------------------------------------------------------
extern "C" void kernel_launch(void* const* d_in, const int* in_sizes, int n_in,
                              void* d_out, int out_size, void* d_ws,
                              size_t ws_size, hipStream_t stream) {
  (void)in_sizes; (void)n_in; (void)out_size; (void)ws_size;
  const float* X   = (const float*)d_in[0];  // [32,2048,256]
  const float* h0  = (const float*)d_in[1];  // [1,32,512]
  const float* Wih = (const float*)d_in[2];  // [512,256]
  const float* Whh = (const float*)d_in[3];  // [512,512]
  const float* bih = (const float*)d_in[4];  // [512]
  const float* bhh = (const float*)d_in[5];  // [512]
  float* Out = (float*)d_out;
  float* wsf = (float*)d_ws;
  unsigned* cnt = (unsigned*)d_ws;

  static bool attr_done = false;
  if (!attr_done) {  // allow >64KB dynamic LDS on the scan kernel
    (void)hipFuncSetAttribute((const void*)rnn_scan,
                              hipFuncAttributeMaxDynamicSharedMemorySize, 165000);
    (void)hipFuncSetAttribute((const void*)rnn_zin_gemm,
                              hipFuncAttributeMaxDynamicSharedMemorySize, 28000);
    attr_done = true;
  }

  // Phase 1: z = X @ Wih^T + b_ih  -> written in place into d_out.
  dim3 g1(H_ / 64, (B_ * S_) / 128);
  size_t smem1 = (size_t)(128 * 36 + 64 * 36) * sizeof(float);
  hipLaunchKernelGGL(rnn_zin_gemm, g1, dim3(256), smem1, stream, X, Wih, bih, Out);

  // Init workspace (counter + h ping-pong buffer 0 = h0), every launch.
  hipLaunchKernelGGL(rnn_init, dim3(64), dim3(256), 0, stream, h0, wsf, cnt);

  // Phase 2: 16 WGs (one max-size cluster's worth), 8 waves each.
  size_t smem2 = (size_t)(2 * 16512 + 8 * 1024) * sizeof(float);
  hipLaunchKernelGGL(rnn_scan, dim3(H_ / 32), dim3(256), smem2, stream, Whh, bhh,
                     Out, wsf, cnt);
}